// VectorQuantize_84301618086202
// MI455X (gfx1250) — compile-verified
//
#include <hip/hip_runtime.h>
#include <hip/hip_bf16.h>

typedef __attribute__((ext_vector_type(2))) float v2f;
typedef __attribute__((ext_vector_type(4))) float v4f;
typedef __attribute__((ext_vector_type(8))) float v8f;

#define D_DIM   1024
#define CD      8
#define CS      1024
#define CHUNK_K 32
#define NCHUNK  (D_DIM / CHUNK_K)   // 32
#define ROWSTR  36                  // padded row stride (16B aligned, bank-conflict-free)

// flat float offsets into d_out (out, commit, cb_loss, indices, z_e)
#define OFF_COMM  67108864LL
#define OFF_CBL   67108880LL
#define OFF_IDX   67108896LL
#define OFF_ZE    67174432LL

// ---------------- prep: weight-norm weights + normalized codebook ----------------
__global__ __launch_bounds__(256) void vq_prep(
    const float* __restrict__ in_v,  const float* __restrict__ in_g,
    const float* __restrict__ out_v, const float* __restrict__ out_g,
    const float* __restrict__ codebook,
    float* __restrict__ Wt,          // [8][1024]  (transposed wn(in_v))
    float* __restrict__ Wout,        // [8][1024]
    float* __restrict__ cbn)         // [1024][8]  l2-normalized codebook
{
    __shared__ float ncol[8];
    const int tid  = threadIdx.x;
    const int lane = tid & 31;
    const int wv   = tid >> 5;
    {
        float s = 0.f;
        for (int d = lane; d < D_DIM; d += 32) { float v = in_v[d * CD + wv]; s += v * v; }
        #pragma unroll
        for (int off = 16; off; off >>= 1) s += __shfl_down(s, off, 32);
        if (lane == 0 && wv < 8) ncol[wv] = sqrtf(s);
    }
    __syncthreads();
    for (int i = tid; i < D_DIM * CD; i += 256) {
        int c = i & 7, d = i >> 3;
        Wt[c * D_DIM + d] = in_g[c] * in_v[i] / fmaxf(1e-12f, ncol[c]);
    }
    for (int d = tid; d < D_DIM; d += 256) {
        float s = 0.f;
        #pragma unroll
        for (int c = 0; c < CD; ++c) { float v = out_v[c * D_DIM + d]; s += v * v; }
        float n = fmaxf(1e-12f, sqrtf(s));
        float g = out_g[d];
        #pragma unroll
        for (int c = 0; c < CD; ++c) Wout[c * D_DIM + d] = g * out_v[c * D_DIM + d] / n;
    }
    for (int r = tid; r < CS; r += 256) {
        float v[8]; float s = 0.f;
        #pragma unroll
        for (int j = 0; j < CD; ++j) { v[j] = codebook[r * CD + j]; s += v[j] * v[j]; }
        float n = fmaxf(1e-12f, sqrtf(s));
        #pragma unroll
        for (int j = 0; j < CD; ++j) cbn[r * CD + j] = v[j] / n;
    }
}

// ---------------- main fused kernel: 256 thr = 8 waves, 16 tokens/wave ----------------
__global__ __launch_bounds__(256) void vq_main(
    const float* __restrict__ z,
    const float* __restrict__ in_b, const float* __restrict__ codebook,
    const float* __restrict__ Wt,   const float* __restrict__ Wout,
    const float* __restrict__ cbn,  const float* __restrict__ out_b,
    float* __restrict__ out, float* __restrict__ partials)
{
    __shared__ __align__(16) float lds_z[2][128 * ROWSTR];  // 36 KB double-buffered z tile
    __shared__ __align__(16) float zq_lds[128 * CD];        // 4 KB
    __shared__ float lds_loss[8];

    const int tid  = threadIdx.x;
    const int lane = tid & 31;
    const int wv   = tid >> 5;
    const int l15  = lane & 15;
    const bool hiL = lane >= 16;
    const int koff = hiL ? 2 : 0;

    const int tok0  = blockIdx.x * 128;
    const int myTok = tok0 + wv * 16 + l15;

    // ---- phase 1: z_e = z @ wn(in_v) + in_b ----
    // Async b128 staging of z[tok0..tok0+127][kb..kb+31] into LDS (ASYNCcnt path),
    // double-buffered; WMMA consumes the B operand from LDS via ds_load_b64.
    const int r_mine   = tid >> 3;
    const int sub_mine = tid & 7;

    // stage chunk c into buffer buf (4 async b128 per thread, 128-row x 128B tile)
    #define STAGE(cc, buf)                                                          \
        {                                                                           \
            const int kb_ = (cc) * CHUNK_K;                                         \
            _Pragma("unroll")                                                       \
            for (int j = 0; j < 4; ++j) {                                           \
                const int r_ = r_mine + 32 * j;                                     \
                const float* gp_ = z + (size_t)(tok0 + r_) * D_DIM + kb_ + sub_mine * 4; \
                unsigned lo_ = (unsigned)(uintptr_t)&lds_z[(buf)][r_ * ROWSTR + sub_mine * 4]; \
                unsigned long long ga_ = (unsigned long long)(uintptr_t)gp_;        \
                asm volatile("global_load_async_to_lds_b128 %0, %1, off"            \
                             :: "v"(lo_), "v"(ga_) : "memory");                     \
            }                                                                       \
        }

    STAGE(0, 0);

    const float* wrowG = Wt + (size_t)(l15 & 7) * D_DIM + koff;   // A operand (W^T), L0-resident
    v8f acc = {0.f, 0.f, 0.f, 0.f, 0.f, 0.f, 0.f, 0.f};

    for (int c = 0; c < NCHUNK; ++c) {
        const int buf = c & 1;
        if (c + 1 < NCHUNK) {
            STAGE(c + 1, buf ^ 1);
            asm volatile("s_wait_asynccnt 0x4" ::: "memory");   // chunk c complete (in-order)
        } else {
            asm volatile("s_wait_asynccnt 0x0" ::: "memory");
        }
        __syncthreads();                                        // publish LDS tile to all waves
        const float* brow = &lds_z[buf][(wv * 16 + l15) * ROWSTR + koff];
        const float* wrow = wrowG + c * CHUNK_K;
        #pragma unroll
        for (int kk = 0; kk < CHUNK_K; kk += 4) {
            v2f a = *(const v2f*)(wrow + kk);
            v2f b = *(const v2f*)(brow + kk);
            acc = __builtin_amdgcn_wmma_f32_16x16x4_f32(false, a, false, b,
                                                        (short)0, acc, false, false);
        }
        __syncthreads();                                        // tile consumed, safe to overwrite
    }
    #pragma unroll
    for (int j = 0; j < 8; ++j) acc[j] += in_b[j];
    // lane<16: acc[j] == z_e[myTok][j]

    // ---- phase 2: nearest code via WMMA (A = normalized codebook tile, B = z_e^T) ----
    float e0 = __shfl(acc[0], l15, 32), e1 = __shfl(acc[1], l15, 32);
    float e2 = __shfl(acc[2], l15, 32), e3 = __shfl(acc[3], l15, 32);
    float e4 = __shfl(acc[4], l15, 32), e5 = __shfl(acc[5], l15, 32);
    float e6 = __shfl(acc[6], l15, 32), e7 = __shfl(acc[7], l15, 32);
    v2f blo, bhi;
    blo.x = hiL ? e2 : e0;  blo.y = hiL ? e3 : e1;
    bhi.x = hiL ? e6 : e4;  bhi.y = hiL ? e7 : e5;

    float best = -3.4e38f;
    int bestIdx = 0;
    for (int t = 0; t < 64; ++t) {
        const float* cr = cbn + (size_t)(t * 16 + l15) * CD + koff;   // L0-resident 32 KB
        v2f alo = *(const v2f*)cr;
        v2f ahi = *(const v2f*)(cr + 4);
        v8f s = {0.f, 0.f, 0.f, 0.f, 0.f, 0.f, 0.f, 0.f};
        s = __builtin_amdgcn_wmma_f32_16x16x4_f32(false, alo, false, blo, (short)0, s, false, false);
        s = __builtin_amdgcn_wmma_f32_16x16x4_f32(false, ahi, false, bhi, (short)0, s, false, false);
        const int bidx = t * 16 + (hiL ? 8 : 0);
        #pragma unroll
        for (int j = 0; j < 8; ++j) {
            float v = s[j];
            if (v > best) { best = v; bestIdx = bidx + j; }
        }
    }
    {   // merge the two half-wave candidate sets (tie -> smaller index, argmax semantics)
        float ob = __shfl_xor(best, 16, 32);
        int   oi = __shfl_xor(bestIdx, 16, 32);
        if (ob > best || (ob == best && oi < bestIdx)) { best = ob; bestIdx = oi; }
    }

    // ---- phase 3: gather z_q, losses, write indices / z_e ----
    const float* cbr = codebook + (size_t)bestIdx * CD;
    v4f q0 = *(const v4f*)cbr;
    v4f q1 = *(const v4f*)(cbr + 4);

    float d0 = acc[0] - q0.x, d1 = acc[1] - q0.y, d2 = acc[2] - q0.z, d3 = acc[3] - q0.w;
    float d4 = acc[4] - q1.x, d5 = acc[5] - q1.y, d6 = acc[6] - q1.z, d7 = acc[7] - q1.w;
    float ls = d0*d0 + d1*d1 + d2*d2 + d3*d3 + d4*d4 + d5*d5 + d6*d6 + d7*d7;
    if (hiL) ls = 0.f;

    if (!hiL) {
        out[OFF_IDX + myTok] = (float)bestIdx;
        v4f t0; t0.x = acc[0]; t0.y = acc[1]; t0.z = acc[2]; t0.w = acc[3];
        v4f t1; t1.x = acc[4]; t1.y = acc[5]; t1.z = acc[6]; t1.w = acc[7];
        *(v4f*)(out + OFF_ZE + (size_t)myTok * CD)     = t0;
        *(v4f*)(out + OFF_ZE + (size_t)myTok * CD + 4) = t1;
        const int r = wv * 16 + l15;
        *(v4f*)(zq_lds + r * CD)     = q0;
        *(v4f*)(zq_lds + r * CD + 4) = q1;
    }
    #pragma unroll
    for (int off = 16; off; off >>= 1) ls += __shfl_down(ls, off, 32);
    if (lane == 0) lds_loss[wv] = ls;
    __syncthreads();                    // also publishes zq_lds for phase 4
    if (tid == 0) {
        float s = 0.f;
        #pragma unroll
        for (int w = 0; w < 8; ++w) s += lds_loss[w];
        partials[blockIdx.x] = s;       // deterministic per-block partial
    }

    // ---- phase 4: out = z_q @ wn(out_v) + out_b (coalesced nontemporal b128 stores) ----
    const int dcol = tid * 4;
    v4f w[8];
    #pragma unroll
    for (int c = 0; c < 8; ++c) w[c] = *(const v4f*)(Wout + c * D_DIM + dcol);
    v4f ob4 = *(const v4f*)(out_b + dcol);
    float* op = out + (size_t)tok0 * D_DIM + dcol;
    for (int t = 0; t < 128; ++t) {
        const float* q = zq_lds + t * CD;
        v4f r = ob4;
        #pragma unroll
        for (int c = 0; c < 8; ++c) {
            float s = q[c];
            r.x += w[c].x * s; r.y += w[c].y * s;
            r.z += w[c].z * s; r.w += w[c].w * s;
        }
        __builtin_nontemporal_store(r, (v4f*)(op + (size_t)t * D_DIM));
    }
}

// ---------------- final deterministic loss reduction ----------------
__global__ void vq_loss(const float* __restrict__ partials, float* __restrict__ out)
{
    const int b = threadIdx.x;
    if (b < 16) {
        float s = 0.f;
        for (int i = 0; i < 32; ++i) s += partials[b * 32 + i];
        s *= (1.0f / (4096.0f * 8.0f));   // mean over (T, CD)
        out[OFF_COMM + b] = s;            // commitment_loss
        out[OFF_CBL  + b] = s;            // codebook_loss (identical forward value)
    }
}

extern "C" void kernel_launch(void* const* d_in, const int* in_sizes, int n_in,
                              void* d_out, int out_size, void* d_ws, size_t ws_size,
                              hipStream_t stream) {
    const float* z     = (const float*)d_in[0];
    const float* in_v  = (const float*)d_in[1];
    const float* in_g  = (const float*)d_in[2];
    const float* in_b  = (const float*)d_in[3];
    const float* out_v = (const float*)d_in[4];
    const float* out_g = (const float*)d_in[5];
    const float* out_b = (const float*)d_in[6];
    const float* cbk   = (const float*)d_in[7];
    float* out = (float*)d_out;
    float* ws  = (float*)d_ws;

    float* Wt       = ws;            // 8192 floats
    float* Wout     = ws + 8192;     // 8192 floats
    float* cbn      = ws + 16384;    // 8192 floats
    float* partials = ws + 24576;    // 512 floats

    hipLaunchKernelGGL(vq_prep, dim3(1),   dim3(256), 0, stream,
                       in_v, in_g, out_v, out_g, cbk, Wt, Wout, cbn);
    hipLaunchKernelGGL(vq_main, dim3(512), dim3(256), 0, stream,
                       z, in_b, cbk, Wt, Wout, cbn, out_b, out, partials);
    hipLaunchKernelGGL(vq_loss, dim3(1),   dim3(32),  0, stream, partials, out);
}